// IoULoss_35862976921868
// MI455X (gfx1250) — compile-verified
//
#include <hip/hip_runtime.h>

typedef __attribute__((ext_vector_type(2))) float v2f;
typedef __attribute__((ext_vector_type(4))) float f4;
typedef __attribute__((ext_vector_type(8))) float v8f;

#define IOU_EPS 1e-7f

// Per-row IoU math. Accumulates sum(log(iou+eps)) and sum(iou).
__device__ __forceinline__ void iou_row(float ox, float oy, float os,
                                        float tx, float ty, float ts,
                                        float& log_acc, float& iou_acc) {
    float oa = 4.0f * os * os;                       // (2*out_size)^2
    float ta = 4.0f * ts * ts;                       // (2*tar_size)^2
    float w  = fminf(ox + os, tx + ts) - fmaxf(ox - os, tx - ts);
    w = fmaxf(w, 0.0f);
    float h  = fminf(oy + os, ty + ts) - fmaxf(oy - os, ty - ts);
    h = fmaxf(h, 0.0f);
    float ov  = w * h;
    float un  = oa + ta - ov;
    float iou = ov / (un + IOU_EPS);
    log_acc += logf(iou + IOU_EPS);
    iou_acc += iou;
}

// Full wave32 reduction of a (loss, iou) pair per lane via one
// V_WMMA_F32_16X16X4_F32.
//   A (16x4): a.x = lane's loss partial (K=0 / K=2), a.y = iou partial (K=1 / K=3)
//   B (4x16): B[k][n] = (k < 2) ? (n<8) : (n>=8)  -> built as
//             b = (lane<8 || lane>=24) ? 1 : 0 in both B VGPRs.
// Result after summing D VGPRs + shfl_xor(16):
//   lane 0 holds loss+iou summed over lanes 0-15, lane 8 over lanes 16-31 (all lanes valid
//   within their column class).
__device__ __forceinline__ float wave_pair_reduce_wmma(float loss_p, float iou_p) {
    int lane = threadIdx.x & 31;
    float bv = ((lane < 8) || (lane >= 24)) ? 1.0f : 0.0f;
    v2f a; a.x = loss_p; a.y = iou_p;
    v2f b; b.x = bv;     b.y = bv;
    v8f c = {};
    v8f d = __builtin_amdgcn_wmma_f32_16x16x4_f32(
        /*neg_a=*/false, a, /*neg_b=*/false, b,
        /*c_mod=*/(short)0, c, /*reuse_a=*/false, /*reuse_b=*/false);
    float t = d[0] + d[1] + d[2] + d[3] + d[4] + d[5] + d[6] + d[7];
    // add the other 8 rows (upper lane half holds rows 8..15)
    t += __shfl_xor(t, 16, 32);
    return t;   // lane 0: loss+iou of lanes 0-15, lane 8: loss+iou of lanes 16-31
}

// Pass 1: streaming IoU + per-block loss+iou partials of wave lanes 0-15 / 16-31 -> ws[2*bid], ws[2*bid+1]
__global__ void __launch_bounds__(256)
iou_partial_kernel(const float* __restrict__ outs, const float* __restrict__ tars,
                   float* __restrict__ ws, int n_rows) {
    __shared__ float s_loss[8];
    __shared__ float s_iou[8];

    const int tid      = blockIdx.x * blockDim.x + threadIdx.x;
    const int nthreads = gridDim.x * blockDim.x;
    const int groups   = n_rows >> 2;   // 4 rows (=12 floats = 3x float4) per step

    float loss_p = 0.0f, iou_p = 0.0f;

    for (int g = tid; g < groups; g += nthreads) {
        const f4* ob = (const f4*)(outs + (size_t)g * 12);
        const f4* tb = (const f4*)(tars + (size_t)g * 12);
        // Non-temporal 128-bit streaming loads: single-use data, keep L2 clean.
        f4 o0 = __builtin_nontemporal_load(ob + 0);
        f4 o1 = __builtin_nontemporal_load(ob + 1);
        f4 o2 = __builtin_nontemporal_load(ob + 2);
        f4 t0 = __builtin_nontemporal_load(tb + 0);
        f4 t1 = __builtin_nontemporal_load(tb + 1);
        f4 t2 = __builtin_nontemporal_load(tb + 2);

        iou_row(o0.x, o0.y, o0.z, t0.x, t0.y, t0.z, loss_p, iou_p);
        iou_row(o0.w, o1.x, o1.y, t0.w, t1.x, t1.y, loss_p, iou_p);
        iou_row(o1.z, o1.w, o2.x, t1.z, t1.w, t2.x, loss_p, iou_p);
        iou_row(o2.y, o2.z, o2.w, t2.y, t2.z, t2.w, loss_p, iou_p);
    }

    // Tail rows (n_rows % 4) handled by global thread 0; reconverges before WMMA.
    if (tid == 0) {
        for (int r = groups << 2; r < n_rows; ++r) {
            iou_row(outs[3 * r], outs[3 * r + 1], outs[3 * r + 2],
                    tars[3 * r], tars[3 * r + 1], tars[3 * r + 2],
                    loss_p, iou_p);
        }
    }

    // Wave-level reduction via WMMA (EXEC all-ones here: full block, post-loop).
    float t = wave_pair_reduce_wmma(loss_p, iou_p);

    const int lane = threadIdx.x & 31;
    const int wave = threadIdx.x >> 5;
    if (lane == 0) s_loss[wave] = t;
    if (lane == 8) s_iou[wave]  = t;
    __syncthreads();

    // Deterministic serial cross-wave reduction (8 waves) by thread 0.
    if (threadIdx.x == 0) {
        float bl = 0.0f, bi = 0.0f;
        #pragma unroll
        for (int w = 0; w < 8; ++w) { bl += s_loss[w]; bi += s_iou[w]; }
        ws[2 * blockIdx.x + 0] = bl;
        ws[2 * blockIdx.x + 1] = bi;
    }
}

// Pass 2: one wave reduces the per-block partials; WMMA epilogue again.
__global__ void __launch_bounds__(32)
iou_final_kernel(const float* __restrict__ ws, float* __restrict__ out, int nblocks) {
    const int lane = threadIdx.x;   // 32 threads = 1 wave
    float loss_p = 0.0f, iou_p = 0.0f;
    for (int b = lane; b < nblocks; b += 32) {
        loss_p += ws[2 * b + 0];
        iou_p  += ws[2 * b + 1];
    }
    float t = wave_pair_reduce_wmma(loss_p, iou_p);
    if (lane == 0) out[0] = -t;   // -(sum log(iou+eps) + sum iou) of blocks b%32 < 16
    if (lane == 8) out[1] = t;    // sum log(iou+eps) + sum iou of blocks b%32 >= 16
}

extern "C" void kernel_launch(void* const* d_in, const int* in_sizes, int n_in,
                              void* d_out, int out_size, void* d_ws, size_t ws_size,
                              hipStream_t stream) {
    const float* outs = (const float*)d_in[0];
    const float* tars = (const float*)d_in[1];
    float* out = (float*)d_out;
    float* ws  = (float*)d_ws;

    const int n_rows = in_sizes[0] / 3;

    int blocks = 2048;                               // 16K waves -> saturate HBM
    size_t need = (size_t)blocks * 2 * sizeof(float);
    if (ws_size < need) {
        blocks = (int)(ws_size / (2 * sizeof(float)));
        if (blocks < 1) blocks = 1;
    }

    iou_partial_kernel<<<blocks, 256, 0, stream>>>(outs, tars, ws, n_rows);
    iou_final_kernel<<<1, 32, 0, stream>>>(ws, out, blocks);
}